// ODEModel_74732430950754
// MI455X (gfx1250) — compile-verified
//
#include <hip/hip_runtime.h>
#include <math.h>

// ---------------- problem / workspace layout ----------------
#define BATCH   262144
#define NB      1024               // BATCH / 256
#define NTOT    (BATCH*6)

#define S_T     0                  // scalar slots (floats) in ws
#define S_H     1
#define S_APPLY 2
#define S_H0    3
#define S_D1    4
#define P0_OFF  64                 // per-block partial sums (NB each)
#define P1_OFF  (P0_OFF + NB)
#define P2_OFF  (P0_OFF + 2*NB)
#define PK_OFF  8192               // packed WMMA constants
#define PK_TOT  4480               // 896 (A1) + 1792 (A2) + 1792 (b1C)
#define ARR_OFF 16384              // 24 planes of BATCH floats: y, fy, ynew, k7

// LDS pack offsets (floats)
#define L_A1  0
#define L_A2  896
#define L_B1C 2688

// Dopri5 tableau (fp32)
#define CA21 0.2f
#define CA31 0.075f
#define CA32 0.225f
#define CA41 ((float)(44.0/45.0))
#define CA42 ((float)(-56.0/15.0))
#define CA43 ((float)(32.0/9.0))
#define CA51 ((float)(19372.0/6561.0))
#define CA52 ((float)(-25360.0/2187.0))
#define CA53 ((float)(64448.0/6561.0))
#define CA54 ((float)(-212.0/729.0))
#define CA61 ((float)(9017.0/3168.0))
#define CA62 ((float)(-355.0/33.0))
#define CA63 ((float)(46732.0/5247.0))
#define CA64 ((float)(49.0/176.0))
#define CA65 ((float)(-5103.0/18656.0))
#define CB1  ((float)(35.0/384.0))
#define CB3  ((float)(500.0/1113.0))
#define CB4  ((float)(125.0/192.0))
#define CB5  ((float)(-2187.0/6784.0))
#define CB6  ((float)(11.0/84.0))
#define CE1  ((float)(71.0/57600.0))
#define CE3  ((float)(-71.0/16695.0))
#define CE4  ((float)(71.0/1920.0))
#define CE5  ((float)(-17253.0/339200.0))
#define CE6  ((float)(22.0/525.0))
#define CE7  (-0.025f)

#define TWO_LOG2E 2.8853900817779268f   // 2*log2(e)

typedef float v2f __attribute__((ext_vector_type(2)));
typedef float v8f __attribute__((ext_vector_type(8)));

// D = A(16x4 f32) * B(4x16 f32) + C(16x16 f32)
#define WMMA4(a,b,c) __builtin_amdgcn_wmma_f32_16x16x4_f32(false,(a),false,(b),(short)0,(c),false,false)

// base-2 exp/log map directly to v_exp_f32 / v_log_f32
__device__ __forceinline__ float dev_exp2(float x){ return __builtin_amdgcn_exp2f(x); }
__device__ __forceinline__ float dev_log2(float x){ return __builtin_amdgcn_logf(x); }

// half-wave swap (lanes l <-> l^16), SWAPX16: and=0x1f, or=0, xor=0x10
__device__ __forceinline__ float swap16(float v){
  return __int_as_float(__builtin_amdgcn_ds_swizzle(__float_as_int(v), 0x401F));
}
__device__ __forceinline__ float fast_tanh(float x){
  float e = dev_exp2(x * TWO_LOG2E);               // e^(2x) via v_exp_f32
  return 1.0f - 2.0f*__builtin_amdgcn_rcpf(e + 1.0f);
}

__device__ __forceinline__ void stage_pack(const float* __restrict__ wsf, float* __restrict__ L){
  for (int i = threadIdx.x; i < PK_TOT; i += blockDim.x) L[i] = wsf[PK_OFF + i];
  __syncthreads();
}

__device__ __forceinline__ float block_reduce256(float* red, float v){
  int tid = threadIdx.x;
  red[tid] = v; __syncthreads();
  #pragma unroll
  for (int s = 128; s > 0; s >>= 1){
    if (tid < s) red[tid] += red[tid + s];
    __syncthreads();
  }
  float r = red[0]; __syncthreads();
  return r;
}

// ---------------- ODE right-hand side via fp32 WMMA ----------------
// State: lane l of a wave holds element (waveBase+l); per-lane y[0..5].
// Zt = W1 @ y^T (per 16-col half-tile), tanh, out^T = W2pad @ Act^T.
__device__ __forceinline__ void ode_f(const float* __restrict__ L, const float* __restrict__ b2v,
                                      const float* __restrict__ y, float* __restrict__ dy,
                                      int lane, bool lo){
  float ys[6];
  #pragma unroll
  for (int d = 0; d < 6; ++d) ys[d] = swap16(y[d]);

  // y^T in B-layout (4x16): v0 = K=(4k+2q), v1 = K=(4k+1+2q), q = lane/16
  v2f bt0k0, bt0k1, bt1k0, bt1k1;
  bt0k0.x = lo ? y[0]  : ys[2];  bt0k0.y = lo ? y[1]  : ys[3];
  bt0k1.x = lo ? y[4]  : 0.0f;   bt0k1.y = lo ? y[5]  : 0.0f;
  bt1k0.x = lo ? ys[0] : y[2];   bt1k0.y = lo ? ys[1] : y[3];
  bt1k1.x = lo ? ys[4] : 0.0f;   bt1k1.y = lo ? ys[5] : 0.0f;

  v8f acc0 = {0,0,0,0,0,0,0,0};
  v8f acc1 = {0,0,0,0,0,0,0,0};

  // Keep this loop ROLLED: full unroll keeps 7 tiles of Z live and spills to scratch.
  #pragma unroll 1
  for (int t = 0; t < 7; ++t){
    const float* bc = L + L_B1C + t*256 + lane*8;   // b1 tile in C-layout (free C operand)
    v8f c;
    #pragma unroll
    for (int r = 0; r < 8; ++r) c[r] = bc[r];
    v2f a0 = *(const v2f*)(L + L_A1 + (t*2+0)*64 + lane*2);
    v2f a1 = *(const v2f*)(L + L_A1 + (t*2+1)*64 + lane*2);

    v8f z0 = WMMA4(a0, bt0k0, c);  z0 = WMMA4(a1, bt0k1, z0);   // Z^T tile, elems 0-15
    v8f z1 = WMMA4(a0, bt1k0, c);  z1 = WMMA4(a1, bt1k1, z1);   // Z^T tile, elems 16-31

    float z0a[8], z1a[8], zs0[8], zs1[8];
    #pragma unroll
    for (int r = 0; r < 8; ++r){
      z0a[r] = fast_tanh(z0[r]);  z1a[r] = fast_tanh(z1[r]);
      zs0[r] = swap16(z0a[r]);    zs1[r] = swap16(z1a[r]);
    }
    // Act^T (C-layout) -> B-layout k-steps of 4; accumulate out^T = W2pad @ Act^T
    #pragma unroll
    for (int j = 0; j < 4; ++j){
      v2f a2 = *(const v2f*)(L + L_A2 + (4*t+j)*64 + lane*2);
      v2f b0, b1;
      if (j < 2){
        b0.x = lo ? z0a[4*j]   : zs0[4*j+2];
        b0.y = lo ? z0a[4*j+1] : zs0[4*j+3];
        b1.x = lo ? z1a[4*j]   : zs1[4*j+2];
        b1.y = lo ? z1a[4*j+1] : zs1[4*j+3];
      } else {
        b0.x = lo ? zs0[4*j-8] : z0a[4*j-6];
        b0.y = lo ? zs0[4*j-7] : z0a[4*j-5];
        b1.x = lo ? zs1[4*j-8] : z1a[4*j-6];
        b1.y = lo ? zs1[4*j-7] : z1a[4*j-5];
      }
      acc0 = WMMA4(a2, b0, acc0);
      acc1 = WMMA4(a2, b1, acc1);
    }
  }
  // out^T C-layout: reg d, lanes 0-15 hold (d, m). Reassemble per-lane dy + b2.
  #pragma unroll
  for (int d = 0; d < 6; ++d){
    float sw = swap16(acc1[d]);
    dy[d] = (lo ? acc0[d] : sw) + b2v[d];
  }
}

// ---------------- kernels ----------------
// Pack W1/W2 into WMMA A-layout and b1 into C-layout tiles (runs once per launch).
__global__ void k_pack(const float* __restrict__ W1, const float* __restrict__ b1,
                       const float* __restrict__ W2, float* __restrict__ wsf){
  for (int i = threadIdx.x; i < PK_TOT; i += blockDim.x){
    float v = 0.0f;
    if (i < 896){                       // A1: W1[h][d], M=h tile row, K=d
      int tk = i/64, r = i%64;
      int t = tk/2, k = tk%2;
      int lane = r/2, vv = r%2;
      int q = lane/16, M = lane%16;
      int d = 4*k + vv + 2*q;
      int h = 16*t + M;
      if (d < 6 && h < 100) v = W1[h*6 + d];
    } else if (i < 2688){               // A2: W2[dd][K], M=dd (pad to 16), K=h (pad to 112)
      int ii = i - 896;
      int jg = ii/64, r = ii%64;
      int lane = r/2, vv = r%2;
      int q = lane/16, dd = lane%16;
      int K = 4*jg + vv + 2*q;
      if (dd < 6 && K < 100) v = W2[dd*100 + K];
    } else {                            // b1 tiles in C-layout
      int ii = i - 2688;
      int t = ii/256, r2 = ii%256;
      int lane = r2/8, r = r2%8;
      int q = lane/16;
      int h = 16*t + r + 8*q;
      if (h < 100) v = b1[h];
    }
    wsf[PK_OFF + i] = v;
  }
}

__global__ void k_init1(const float* __restrict__ seq, const float* __restrict__ b2,
                        float* __restrict__ wsf){
  __shared__ float L[PK_TOT];
  __shared__ float red[256];
  stage_pack(wsf, L);
  float b2v[6];
  #pragma unroll
  for (int d = 0; d < 6; ++d) b2v[d] = b2[d];
  int e = blockIdx.x*blockDim.x + threadIdx.x;
  int lane = threadIdx.x & 31; bool lo = lane < 16;
  float y0[6], f0[6];
  #pragma unroll
  for (int d = 0; d < 6; ++d) y0[d] = seq[e*48 + 42 + d];   // input_seq[:, -1, :]
  ode_f(L, b2v, y0, f0, lane, lo);
  float s0 = 0.f, s1 = 0.f;
  #pragma unroll
  for (int d = 0; d < 6; ++d){
    float sc = 1e-9f + 1e-7f*fabsf(y0[d]);
    float a = y0[d]/sc, b = f0[d]/sc;
    s0 += a*a; s1 += b*b;
    wsf[ARR_OFF + d*BATCH + e]       = y0[d];
    wsf[ARR_OFF + (6+d)*BATCH + e]   = f0[d];
  }
  float r0 = block_reduce256(red, s0);
  float r1 = block_reduce256(red, s1);
  if (threadIdx.x == 0){ wsf[P0_OFF+blockIdx.x] = r0; wsf[P1_OFF+blockIdx.x] = r1; }
}

__global__ void k_init2(float* __restrict__ wsf){
  __shared__ float red[256];
  int tid = threadIdx.x;
  float v0 = wsf[P0_OFF+tid]+wsf[P0_OFF+tid+256]+wsf[P0_OFF+tid+512]+wsf[P0_OFF+tid+768];
  float s0 = block_reduce256(red, v0);
  float v1 = wsf[P1_OFF+tid]+wsf[P1_OFF+tid+256]+wsf[P1_OFF+tid+512]+wsf[P1_OFF+tid+768];
  float s1 = block_reduce256(red, v1);
  if (tid == 0){
    const float invN = 1.0f/(float)NTOT;
    float d0 = sqrtf(s0*invN), d1 = sqrtf(s1*invN);
    float h0 = ((d0 < 1e-5f) || (d1 < 1e-5f)) ? 1e-6f : 0.01f*d0/fmaxf(d1, 1e-12f);
    wsf[S_H0] = h0; wsf[S_D1] = d1;
  }
}

__global__ void k_init3(const float* __restrict__ b2, float* __restrict__ wsf){
  __shared__ float L[PK_TOT];
  __shared__ float red[256];
  stage_pack(wsf, L);
  float b2v[6];
  #pragma unroll
  for (int d = 0; d < 6; ++d) b2v[d] = b2[d];
  int e = blockIdx.x*blockDim.x + threadIdx.x;
  int lane = threadIdx.x & 31; bool lo = lane < 16;
  float h0 = wsf[S_H0];
  float y0[6], f0[6], y1[6], f1[6];
  #pragma unroll
  for (int d = 0; d < 6; ++d){
    y0[d] = wsf[ARR_OFF + d*BATCH + e];
    f0[d] = wsf[ARR_OFF + (6+d)*BATCH + e];
    y1[d] = y0[d] + h0*f0[d];
  }
  ode_f(L, b2v, y1, f1, lane, lo);
  float s2 = 0.f;
  #pragma unroll
  for (int d = 0; d < 6; ++d){
    float sc = 1e-9f + 1e-7f*fabsf(y0[d]);
    float df = (f1[d]-f0[d])/sc;
    s2 += df*df;
  }
  float r2 = block_reduce256(red, s2);
  if (threadIdx.x == 0) wsf[P2_OFF+blockIdx.x] = r2;
}

__global__ void k_init4(float* __restrict__ wsf){
  __shared__ float red[256];
  int tid = threadIdx.x;
  float v2 = wsf[P2_OFF+tid]+wsf[P2_OFF+tid+256]+wsf[P2_OFF+tid+512]+wsf[P2_OFF+tid+768];
  float s2 = block_reduce256(red, v2);
  if (tid == 0){
    float h0 = wsf[S_H0], d1 = wsf[S_D1];
    float d2 = sqrtf(s2/(float)NTOT)/h0;
    float dm = fmaxf(d1, d2);
    float h1 = (dm <= 1e-15f) ? fmaxf(1e-6f, h0*1e-3f)
                              : dev_exp2(0.2f*dev_log2(0.01f/fmaxf(dm, 1e-15f)));
    float h = fminf(fminf(100.0f*h0, h1), 1.0f);
    wsf[S_T] = 0.0f; wsf[S_H] = h; wsf[S_APPLY] = 0.0f;
  }
}

__global__ void k_step(const float* __restrict__ b2, float* __restrict__ wsf){
  __shared__ float L[PK_TOT];
  __shared__ float red[256];
  stage_pack(wsf, L);
  float b2v[6];
  #pragma unroll
  for (int d = 0; d < 6; ++d) b2v[d] = b2[d];
  const int e = blockIdx.x*blockDim.x + threadIdx.x;
  const int lane = threadIdx.x & 31; const bool lo = lane < 16;
  const float t = wsf[S_T];
  const float h = wsf[S_H];
  const bool apply = (wsf[S_APPLY] != 0.0f);

  float y[6], fy[6];
  #pragma unroll
  for (int d = 0; d < 6; ++d){
    float yv  = wsf[ARR_OFF + d*BATCH + e];
    float fv  = wsf[ARR_OFF + (6+d)*BATCH + e];
    float ynv = wsf[ARR_OFF + (12+d)*BATCH + e];
    float k7v = wsf[ARR_OFF + (18+d)*BATCH + e];
    y[d]  = apply ? ynv : yv;     // apply previous step's accept lazily
    fy[d] = apply ? k7v : fv;
  }
  const float hc = fminf(h, 1.0f - t);
  const float* k1 = fy;
  float tmp[6], k2[6], k3[6], k4[6], k5[6], k6[6], yn[6], k7[6];

  #pragma unroll
  for (int d = 0; d < 6; ++d) tmp[d] = y[d] + hc*(CA21*k1[d]);
  ode_f(L, b2v, tmp, k2, lane, lo);
  #pragma unroll
  for (int d = 0; d < 6; ++d) tmp[d] = y[d] + hc*(CA31*k1[d] + CA32*k2[d]);
  ode_f(L, b2v, tmp, k3, lane, lo);
  #pragma unroll
  for (int d = 0; d < 6; ++d) tmp[d] = y[d] + hc*(CA41*k1[d] + CA42*k2[d] + CA43*k3[d]);
  ode_f(L, b2v, tmp, k4, lane, lo);
  #pragma unroll
  for (int d = 0; d < 6; ++d) tmp[d] = y[d] + hc*(CA51*k1[d] + CA52*k2[d] + CA53*k3[d] + CA54*k4[d]);
  ode_f(L, b2v, tmp, k5, lane, lo);
  #pragma unroll
  for (int d = 0; d < 6; ++d) tmp[d] = y[d] + hc*(CA61*k1[d] + CA62*k2[d] + CA63*k3[d] + CA64*k4[d] + CA65*k5[d]);
  ode_f(L, b2v, tmp, k6, lane, lo);
  #pragma unroll
  for (int d = 0; d < 6; ++d) yn[d] = y[d] + hc*(CB1*k1[d] + CB3*k3[d] + CB4*k4[d] + CB5*k5[d] + CB6*k6[d]);
  ode_f(L, b2v, yn, k7, lane, lo);

  float s = 0.f;
  #pragma unroll
  for (int d = 0; d < 6; ++d){
    float err = hc*(CE1*k1[d] + CE3*k3[d] + CE4*k4[d] + CE5*k5[d] + CE6*k6[d] + CE7*k7[d]);
    float sc  = 1e-9f + 1e-7f*fmaxf(fabsf(y[d]), fabsf(yn[d]));
    float r   = err/sc;
    s += r*r;
    wsf[ARR_OFF + d*BATCH + e]       = y[d];    // canonical state
    wsf[ARR_OFF + (6+d)*BATCH + e]   = fy[d];
    wsf[ARR_OFF + (12+d)*BATCH + e]  = yn[d];   // candidates
    wsf[ARR_OFF + (18+d)*BATCH + e]  = k7[d];
  }
  float r = block_reduce256(red, s);
  if (threadIdx.x == 0) wsf[P0_OFF + blockIdx.x] = r;
}

__global__ void k_ctrl(float* __restrict__ wsf){
  __shared__ float red[256];
  int tid = threadIdx.x;
  float v = wsf[P0_OFF+tid]+wsf[P0_OFF+tid+256]+wsf[P0_OFF+tid+512]+wsf[P0_OFF+tid+768];
  float s = block_reduce256(red, v);
  if (tid == 0){
    float en = sqrtf(s/(float)NTOT);
    float t = wsf[S_T], h = wsf[S_H];
    bool done = (t >= 1.0f - 1e-12f);
    float hc = fminf(h, 1.0f - t);
    bool accept = (en <= 1.0f);
    float enc = fmaxf(en, 1e-10f);
    float factor = 0.9f*dev_exp2(-0.2f*dev_log2(enc));
    factor = fminf(fmaxf(factor, 0.2f), 10.0f);
    if (!done){
      wsf[S_T] = accept ? (t + hc) : t;
      wsf[S_H] = hc*factor;
    }
    wsf[S_APPLY] = (accept && !done) ? 1.0f : 0.0f;
  }
}

__global__ void k_out(const float* __restrict__ Wl, const float* __restrict__ bl,
                      const float* __restrict__ wsf, float* __restrict__ out){
  int e = blockIdx.x*blockDim.x + threadIdx.x;
  bool apply = (wsf[S_APPLY] != 0.0f);
  float acc = bl[0];
  #pragma unroll
  for (int d = 0; d < 6; ++d){
    float yv = wsf[ARR_OFF + d*BATCH + e];
    float yn = wsf[ARR_OFF + (12+d)*BATCH + e];
    acc += (apply ? yn : yv) * Wl[d];
  }
  out[e] = acc;
}

extern "C" void kernel_launch(void* const* d_in, const int* in_sizes, int n_in,
                              void* d_out, int out_size, void* d_ws, size_t ws_size,
                              hipStream_t stream) {
  (void)in_sizes; (void)n_in; (void)out_size; (void)ws_size;
  const float* seq = (const float*)d_in[0];
  const float* W1  = (const float*)d_in[1];
  const float* b1  = (const float*)d_in[2];
  const float* W2  = (const float*)d_in[3];
  const float* b2  = (const float*)d_in[4];
  const float* Wl  = (const float*)d_in[5];
  const float* bl  = (const float*)d_in[6];
  float* wsf = (float*)d_ws;
  float* out = (float*)d_out;

  k_pack <<<1,   256, 0, stream>>>(W1, b1, W2, wsf);
  k_init1<<<NB,  256, 0, stream>>>(seq, b2, wsf);
  k_init2<<<1,   256, 0, stream>>>(wsf);
  k_init3<<<NB,  256, 0, stream>>>(b2, wsf);
  k_init4<<<1,   256, 0, stream>>>(wsf);
  for (int s = 0; s < 64; ++s){
    k_step<<<NB, 256, 0, stream>>>(b2, wsf);
    k_ctrl<<<1,  256, 0, stream>>>(wsf);
  }
  k_out  <<<NB,  256, 0, stream>>>(Wl, bl, wsf, out);
}